// SpectralContrastPrior_62672162783631
// MI455X (gfx1250) — compile-verified
//
#include <hip/hip_runtime.h>

// SpectralContrastPrior: per-(batch,band,t) top-3 / bottom-3 mean over the
// frequency axis. Streaming, bandwidth-bound: 262 MB read once -> ~11.3 us
// floor at 23.3 TB/s. No matmul structure -> WMMA inapplicable; we use the
// CDNA5 async global->LDS DMA path for deep memory pipelining instead.

#ifndef __has_builtin
#define __has_builtin(x) 0
#endif

#define N_BANDS 6
#define NFREQ   1025
#define TLEN    2000
#define NBATCH  32
#define BANDSZ  170                 // NFREQ / N_BANDS
#define LASTSZ  175                 // NFREQ - 5*BANDSZ
#define VEC     4                   // float4 per thread
#define NVCOL   (TLEN / VEC)        // 500 vector columns
#define BLOCK   128
#define NS      16                  // async pipeline depth (rows in flight/wave)

#if defined(__HIP_DEVICE_COMPILE__) && \
    __has_builtin(__builtin_amdgcn_global_load_async_to_lds_b128) && \
    __has_builtin(__builtin_amdgcn_s_wait_asynccnt)
#define ASYNC_OK 1
#else
#define ASYNC_OK 0
#endif

__device__ __forceinline__ float fmed3(float a, float b, float c) {
#if __has_builtin(__builtin_amdgcn_fmed3f)
  return __builtin_amdgcn_fmed3f(a, b, c);   // v_med3_num_f32
#else
  return fmaxf(fminf(fmaxf(a, b), c), fminf(a, b));
#endif
}

// Insert x into running top-3 (a1>=a2>=a3) and bottom-3 (z1<=z2<=z3).
// Order matters: a3 uses old a2, a2 uses old a1 (same for z side).
#define UPD1(x, a1, a2, a3, z1, z2, z3) do {   \
    float _x = (x);                            \
    (a3) = fmed3(_x, (a2), (a3));              \
    (a2) = fmed3(_x, (a1), (a2));              \
    (a1) = fmaxf(_x, (a1));                    \
    (z3) = fmed3(_x, (z2), (z3));              \
    (z2) = fmed3(_x, (z1), (z2));              \
    (z1) = fminf(_x, (z1));                    \
  } while (0)

#define UPDV(v) do {                                      \
    UPD1((v).x, a1.x, a2.x, a3.x, z1.x, z2.x, z3.x);      \
    UPD1((v).y, a1.y, a2.y, a3.y, z1.y, z2.y, z3.y);      \
    UPD1((v).z, a1.z, a2.z, a3.z, z1.z, z2.z, z3.z);      \
    UPD1((v).w, a1.w, a2.w, a3.w, z1.w, z2.w, z3.w);      \
  } while (0)

#if ASYNC_OK
// The builtin's parameter types (from clang's diagnostic) are:
//   (int4-vector __device__(AS1) *, int4-vector __shared__(AS3) *, int, int)
typedef int v4i_t __attribute__((vector_size(16)));
typedef __attribute__((address_space(1))) v4i_t* g_v4i_p;   // global
typedef __attribute__((address_space(3))) v4i_t* l_v4i_p;   // LDS

__device__ __forceinline__ void asyncCopy16(const float* g, float4* l) {
  // GLOBAL_LOAD_ASYNC_TO_LDS_B128: per-lane 16B global -> LDS, ASYNCcnt-tracked.
  __builtin_amdgcn_global_load_async_to_lds_b128(
      (g_v4i_p)(g), (l_v4i_p)(l), 0, 0);
}
#endif

__global__ __launch_bounds__(BLOCK, 1)
void spectral_contrast_kernel(const float* __restrict__ in,
                              float* __restrict__ out) {
  const int tid  = (int)threadIdx.x;
  const int band = (int)blockIdx.y;
  const int b    = (int)blockIdx.z;
  const int c    = (int)blockIdx.x * BLOCK + tid;   // vector column index in t

#if ASYNC_OK
  __shared__ float4 stage[NS][BLOCK];   // 16 * 128 * 16B = 32 KB (of 320 KB/WGP)
#endif

  if (c >= NVCOL) return;

  const int f0    = band * BANDSZ;
  const int nrows = (band == N_BANDS - 1) ? LASTSZ : BANDSZ;   // 170 or 175
  const float* rowp =
      in + ((size_t)b * NFREQ + (size_t)f0) * TLEN + (size_t)c * VEC;

  const float ninf = -__builtin_inff();
  const float pinf =  __builtin_inff();
  float4 a1, a2, a3, z1, z2, z3;
  a1.x = a1.y = a1.z = a1.w = ninf; a2 = a1; a3 = a1;
  z1.x = z1.y = z1.z = z1.w = pinf; z2 = z1; z3 = z1;

#if ASYNC_OK
  // ---- 16-deep async global->LDS pipeline: 512 B per wave per stage ----
  #pragma unroll
  for (int r = 0; r < NS; ++r)
    asyncCopy16(rowp + (size_t)r * TLEN, &stage[r][tid]);

  for (int r = 0; r < nrows - NS; ++r) {
    __builtin_amdgcn_s_wait_asynccnt(NS - 1);    // oldest stage landed (in-order)
    float4 v = stage[r & (NS - 1)][tid];
    UPDV(v);
    // Ensure the ds_load above fully retired before the async engine may
    // rewrite this slot (ASYNC ops are unordered vs LDS ops).
    asm volatile("s_wait_dscnt 0x0" ::: "memory");
    asyncCopy16(rowp + (size_t)(r + NS) * TLEN, &stage[r & (NS - 1)][tid]);
  }

  // Drain: wait immediates must be compile-time constants.
  #define EPI(J) {                                              \
    __builtin_amdgcn_s_wait_asynccnt(NS - 1 - (J));             \
    float4 v = stage[(nrows - NS + (J)) & (NS - 1)][tid];       \
    UPDV(v); }
  EPI(0)  EPI(1)  EPI(2)  EPI(3)  EPI(4)  EPI(5)  EPI(6)  EPI(7)
  EPI(8)  EPI(9)  EPI(10) EPI(11) EPI(12) EPI(13) EPI(14) EPI(15)
  #undef EPI
#else
  // ---- fallback: direct b128 loads, unrolled for memory-level parallelism ----
  #pragma unroll 8
  for (int r = 0; r < nrows; ++r) {
    float4 v = *(const float4*)(rowp + (size_t)r * TLEN);
    UPDV(v);
  }
#endif

  float4 pk, vl;
  pk.x = (a1.x + a2.x + a3.x) / 3.0f;
  pk.y = (a1.y + a2.y + a3.y) / 3.0f;
  pk.z = (a1.z + a2.z + a3.z) / 3.0f;
  pk.w = (a1.w + a2.w + a3.w) / 3.0f;
  vl.x = (z1.x + z2.x + z3.x) / 3.0f;
  vl.y = (z1.y + z2.y + z3.y) / 3.0f;
  vl.z = (z1.z + z2.z + z3.z) / 3.0f;
  vl.w = (z1.w + z2.w + z3.w) / 3.0f;

  const size_t oidx = ((size_t)b * N_BANDS + band) * TLEN + (size_t)c * VEC;
  *(float4*)(out + oidx) = pk;                                          // peaks
  *(float4*)(out + (size_t)NBATCH * N_BANDS * TLEN + oidx) = vl;        // valleys
}

extern "C" void kernel_launch(void* const* d_in, const int* in_sizes, int n_in,
                              void* d_out, int out_size, void* d_ws, size_t ws_size,
                              hipStream_t stream) {
  (void)in_sizes; (void)n_in; (void)out_size; (void)d_ws; (void)ws_size;
  const float* in  = (const float*)d_in[0];
  float*       out = (float*)d_out;
  dim3 grid((NVCOL + BLOCK - 1) / BLOCK, N_BANDS, NBATCH);   // (4, 6, 32)
  spectral_contrast_kernel<<<grid, dim3(BLOCK), 0, stream>>>(in, out);
}